// HDLoss_11381663334571
// MI455X (gfx1250) — compile-verified
//
#include <hip/hip_runtime.h>
#include <hip/hip_bf16.h>
#include <math.h>

typedef float v2f __attribute__((ext_vector_type(2)));
typedef float v8f __attribute__((ext_vector_type(8)));

// Directed min kernel: for each row n of X (16 rows per wave), compute
// rowmin[n] = min_m ( |x_n|^2 + |y_m|^2 - 2 x_n . y_m ).
// Since |x_n|^2 is constant over m, the loop accumulates min_m(|y_m|^2 - 2 x.y)
// and adds |x_n|^2 once afterwards -> smaller VALU epilogue per WMMA tile.
// One V_WMMA_F32_16X16X4_F32 per 16x16 tile; D=3 zero-padded to K=4.
__global__ __launch_bounds__(256)
void hd_dirmin_kernel(const float* __restrict__ X, const float* __restrict__ Y,
                      float* __restrict__ rowmin, int nrows, int ncols) {
    const int lane = threadIdx.x & 31;
    const int wave = threadIdx.x >> 5;
    const int wave_global = blockIdx.x * (blockDim.x >> 5) + wave;

    const int tilesPerBatch = nrows >> 4;
    const int b = wave_global / tilesPerBatch;
    const int ntile = wave_global - b * tilesPerBatch;
    const int n_base = ntile << 4;

    const float* Xb = X + (size_t)b * (size_t)nrows * 3;
    const float* Yb = Y + (size_t)b * (size_t)ncols * 3;

    const int half = lane >> 4;   // 0: K=0,1  1: K=2,3(pad)
    const int l16  = lane & 15;

    // ---- A operand (16x4 fp32): row = n_base + l16, K = 2*half + {0,1} ----
    const float* xr = Xb + (size_t)(n_base + l16) * 3;
    v2f a;
    a.x = half ? xr[2] : xr[0];
    a.y = half ? 0.0f  : xr[1];

    // ---- |x|^2 per C row held by this lane: row = n_base + v + 8*half ----
    float x2v[8];
#pragma unroll
    for (int v = 0; v < 8; ++v) {
        const float* p = Xb + (size_t)(n_base + v + 8 * half) * 3;
        x2v[v] = p[0] * p[0] + p[1] * p[1] + p[2] * p[2];
    }

    float rmin[8];
#pragma unroll
    for (int v = 0; v < 8; ++v) rmin[v] = 3.4e38f;

    const int mtiles = ncols >> 4;
#pragma unroll 2
    for (int t = 0; t < mtiles; ++t) {
        const int m = (t << 4) + l16;          // column handled by this lane
        const float* yr = Yb + (size_t)m * 3;
        const float y0 = yr[0], y1 = yr[1], y2c = yr[2];
        const float y2 = y0 * y0 + y1 * y1 + y2c * y2c;

        // ---- B operand (4x16 fp32): col = l16, K = 2*half + {0,1} ----
        v2f bv;
        bv.x = half ? y2c  : y0;
        bv.y = half ? 0.0f : y1;

        v8f c = {};
        // D = A x B (+0): 256 dot products x_n . y_m
        c = __builtin_amdgcn_wmma_f32_16x16x4_f32(
                /*neg_a=*/false, a, /*neg_b=*/false, bv,
                /*c_mod=*/(short)0, c, /*reuse_a=*/false, /*reuse_b=*/false);

#pragma unroll
        for (int v = 0; v < 8; ++v) {
            // partial squared distance: |y_m|^2 - 2 x_n . y_m
            rmin[v] = fminf(rmin[v], __builtin_fmaf(-2.0f, c[v], y2));
        }
    }

    // Add the row-constant |x_n|^2 term once, then min across the 16 lanes
    // of each half (each half owns distinct rows).
#pragma unroll
    for (int v = 0; v < 8; ++v) {
        float r = rmin[v] + x2v[v];
        r = fminf(r, __shfl_xor(r, 1, 32));
        r = fminf(r, __shfl_xor(r, 2, 32));
        r = fminf(r, __shfl_xor(r, 4, 32));
        r = fminf(r, __shfl_xor(r, 8, 32));
        rmin[v] = r;
    }
    if (l16 == 0) {
#pragma unroll
        for (int v = 0; v < 8; ++v)
            rowmin[(size_t)b * nrows + n_base + v + 8 * half] = rmin[v];
    }
}

// Final per-batch reduction: out[b] = sqrt(max(0, max(max_n minA, max_m minB)))
__global__ __launch_bounds__(256)
void hd_final_kernel(const float* __restrict__ minA, const float* __restrict__ minB,
                     float* __restrict__ out, int n, int m) {
    const int b = blockIdx.x;
    float mx = 0.0f;  // clamp-to-zero folded into the max identity
    for (int i = threadIdx.x; i < n; i += blockDim.x)
        mx = fmaxf(mx, minA[(size_t)b * n + i]);
    for (int i = threadIdx.x; i < m; i += blockDim.x)
        mx = fmaxf(mx, minB[(size_t)b * m + i]);

    for (int off = 1; off < 32; off <<= 1)
        mx = fmaxf(mx, __shfl_xor(mx, off, 32));

    __shared__ float smax[8];
    const int wave = threadIdx.x >> 5, lane = threadIdx.x & 31;
    if (lane == 0) smax[wave] = mx;
    __syncthreads();
    if (threadIdx.x == 0) {
        float r = smax[0];
        for (int w = 1; w < (int)(blockDim.x >> 5); ++w) r = fmaxf(r, smax[w]);
        out[b] = sqrtf(r);
    }
}

extern "C" void kernel_launch(void* const* d_in, const int* in_sizes, int n_in,
                              void* d_out, int out_size, void* d_ws, size_t ws_size,
                              hipStream_t stream) {
    const float* pred = (const float*)d_in[0];  // [B, N, 3]
    const float* gt   = (const float*)d_in[1];  // [B, M, 3]
    float* out = (float*)d_out;                 // [B]

    const int B = out_size;                     // 4
    const int N = in_sizes[0] / (B * 3);        // 8192
    const int M = in_sizes[1] / (B * 3);        // 8192

    float* minA = (float*)d_ws;                 // [B, N] min over m of d^2
    float* minB = minA + (size_t)B * N;         // [B, M] min over n of d^2

    const int wavesPerBlock = 8;                // 256 threads = 8 wave32
    const int wavesXY = B * (N / 16);
    const int wavesYX = B * (M / 16);

    hd_dirmin_kernel<<<(wavesXY + wavesPerBlock - 1) / wavesPerBlock, 256, 0, stream>>>(
        pred, gt, minA, N, M);
    hd_dirmin_kernel<<<(wavesYX + wavesPerBlock - 1) / wavesPerBlock, 256, 0, stream>>>(
        gt, pred, minB, M, N);
    hd_final_kernel<<<B, 256, 0, stream>>>(minA, minB, out, N, M);
}